// BoundaryLoss_71356586655851
// MI455X (gfx1250) — compile-verified
//
#include <hip/hip_runtime.h>

// Problem shape (fixed by the reference's setup_inputs)
#define BB 8
#define CC 19
#define HH 512
#define WW 1024
#define ROWS (BB * HH)          // 4096
#define BIG  (WW + 1)           // sentinel, matches reference edt_1d_rows

typedef __attribute__((ext_vector_type(2))) float v2f;
typedef __attribute__((ext_vector_type(8))) float v8f;

__device__ __forceinline__ int imax2(int a, int b) { return a > b ? a : b; }
__device__ __forceinline__ int imin2(int a, int b) { return a < b ? a : b; }

// Sum 64 floats held as A (2 VGPRs x 32 lanes) using one V_WMMA_F32_16X16X4_F32
// with B == ones: D[m,n] = sum_k A[m,k] (row sums, replicated across columns).
// Column 0 of D = VGPRs 0..7 on lane 0 (rows 0-7) and lane 16 (rows 8-15), and
// since every column is identical, every lane 0-15 holds rows 0-7 and every
// lane 16-31 holds rows 8-15. Total = shfl(rowblock,0) + shfl(rowblock,16).
// EXEC must be all-1s across the calling wave.
__device__ __forceinline__ float wave_wmma_sum64(float a0, float a1) {
  v2f a; a.x = a0; a.y = a1;
  v2f b; b.x = 1.0f; b.y = 1.0f;
  v8f c = {};
  v8f d = __builtin_amdgcn_wmma_f32_16x16x4_f32(
      /*neg_a=*/false, a, /*neg_b=*/false, b,
      /*c_mod=*/(short)0, c, /*reuse_a=*/false, /*reuse_b=*/false);
  float s = d[0] + d[1] + d[2] + d[3] + d[4] + d[5] + d[6] + d[7];
  return __shfl(s, 0, 32) + __shfl(s, 16, 32);
}

// One block per (b,h) row. 256 threads, 4 pixels each (W = 1024).
__global__ __launch_bounds__(256) void boundary_row_kernel(
    const float* __restrict__ logits,
    const int*   __restrict__ targets,
    float*       __restrict__ partial) {
  __shared__ int   s_last[256];   // inclusive prefix-max of last-nz index per chunk
  __shared__ int   s_first[256];  // inclusive suffix-min of first-nz index per chunk
  __shared__ float s_red[256];

  const int t   = threadIdx.x;
  const int row = blockIdx.x;          // 0 .. B*H-1
  const int b   = row / HH;
  const int h   = row - b * HH;
  const int w0  = t * 4;

  // ---- targets: 4 consecutive pixels, one int4 load ----
  const int4 tg = *reinterpret_cast<const int4*>(targets + (size_t)row * WW + w0);

  // chunk aggregates for the scans
  int last = -BIG;                      // index of last nonzero in chunk
  if (tg.x != 0) last = w0 + 0;
  if (tg.y != 0) last = w0 + 1;
  if (tg.z != 0) last = w0 + 2;
  if (tg.w != 0) last = w0 + 3;
  int first = 2 * BIG;                  // index of first nonzero in chunk
  if (tg.w != 0) first = w0 + 3;
  if (tg.z != 0) first = w0 + 2;
  if (tg.y != 0) first = w0 + 1;
  if (tg.x != 0) first = w0 + 0;

  s_last[t]  = last;
  s_first[t] = first;
  __syncthreads();

  // Hillis-Steele: inclusive prefix-max (last) and inclusive suffix-min (first)
  #pragma unroll
  for (int off = 1; off < 256; off <<= 1) {
    const int lo = (t >= off)       ? s_last[t - off]  : -BIG;
    const int fo = (t + off < 256)  ? s_first[t + off] : 2 * BIG;
    const int lm = s_last[t];
    const int fm = s_first[t];
    __syncthreads();
    s_last[t]  = imax2(lm, lo);
    s_first[t] = imin2(fm, fo);
    __syncthreads();
  }

  int leftC  = (t > 0)   ? s_last[t - 1]  : -BIG;     // exclusive prefix-max
  int rightC = (t < 255) ? s_first[t + 1] : 2 * BIG;  // exclusive suffix-min

  // per-pixel nearest-nonzero left/right (exact 1-D EDT)
  int L0, L1, L2, L3, R0, R1, R2, R3;
  if (tg.x != 0) leftC = w0 + 0;  L0 = leftC;
  if (tg.y != 0) leftC = w0 + 1;  L1 = leftC;
  if (tg.z != 0) leftC = w0 + 2;  L2 = leftC;
  if (tg.w != 0) leftC = w0 + 3;  L3 = leftC;
  if (tg.w != 0) rightC = w0 + 3; R3 = rightC;
  if (tg.z != 0) rightC = w0 + 2; R2 = rightC;
  if (tg.y != 0) rightC = w0 + 1; R1 = rightC;
  if (tg.x != 0) rightC = w0 + 0; R0 = rightC;

  const int d0 = imin2((w0 + 0) - L0, R0 - (w0 + 0));
  const int d1 = imin2((w0 + 1) - L1, R1 - (w0 + 1));
  const int d2 = imin2((w0 + 2) - L2, R2 - (w0 + 2));
  const int d3 = imin2((w0 + 3) - L3, R3 - (w0 + 3));

  // ---- argmax over 19 channels, float4 per channel (coalesced b128) ----
  const float*  lbase   = logits + ((size_t)b * CC * HH + h) * WW + w0;
  const size_t  cstride = (size_t)HH * WW;
  float4 best = *reinterpret_cast<const float4*>(lbase);
  int4   bi   = make_int4(0, 0, 0, 0);
  #pragma unroll
  for (int c = 1; c < CC; ++c) {
    const float4 v = *reinterpret_cast<const float4*>(lbase + (size_t)c * cstride);
    if (v.x > best.x) { best.x = v.x; bi.x = c; }   // strict '>' == first-max,
    if (v.y > best.y) { best.y = v.y; bi.y = c; }   // matches jnp.argmax
    if (v.z > best.z) { best.z = v.z; bi.z = c; }
    if (v.w > best.w) { best.w = v.w; bi.w = c; }
  }

  float local = 0.0f;
  if (bi.x != tg.x) local += (float)d0;
  if (bi.y != tg.y) local += (float)d1;
  if (bi.z != tg.z) local += (float)d2;
  if (bi.w != tg.w) local += (float)d3;

  // ---- block reduction: LDS tree 256 -> 64, then one WMMA ----
  s_red[t] = local;
  __syncthreads();
  if (t < 128) s_red[t] += s_red[t + 128];
  __syncthreads();
  if (t < 64)  s_red[t] += s_red[t + 64];
  __syncthreads();
  if (t < 32) {                                   // wave 0, full EXEC
    const float total = wave_wmma_sum64(s_red[2 * t], s_red[2 * t + 1]);
    if (t == 0) partial[row] = total;
  }
}

// Single wave: fold ROWS partials into 64 lane-accumulators (fixed order),
// finish with one WMMA, scale, and write the scalar result.
__global__ __launch_bounds__(32) void boundary_final_kernel(
    const float* __restrict__ partial,
    float*       __restrict__ out,
    int n, float scale) {
  const int lane = threadIdx.x;
  float a0 = 0.0f, a1 = 0.0f;
  for (int i = 2 * lane; i < n; i += 64) {
    a0 += partial[i];
    a1 += partial[i + 1];
  }
  const float total = wave_wmma_sum64(a0, a1);
  if (lane == 0) out[0] = total * scale;
}

extern "C" void kernel_launch(void* const* d_in, const int* in_sizes, int n_in,
                              void* d_out, int out_size, void* d_ws, size_t ws_size,
                              hipStream_t stream) {
  const float* logits  = (const float*)d_in[0];
  const int*   targets = (const int*)d_in[1];
  float*       partial = (float*)d_ws;       // ROWS floats = 16 KB scratch
  float*       out     = (float*)d_out;

  boundary_row_kernel<<<ROWS, 256, 0, stream>>>(logits, targets, partial);
  boundary_final_kernel<<<1, 32, 0, stream>>>(
      partial, out, ROWS, 1.0f / (float)((size_t)BB * HH * WW));
}